// MagnetLossEval_75179107549356
// MI455X (gfx1250) — compile-verified
//
#include <hip/hip_runtime.h>
#include <hip/hip_bf16.h>

// MagnetLoss eval for MI455X (gfx1250, wave32).
// costs = ||m_c - x||^2 over C=262144 prototypes, D=128; exact top-128 via
// u64 (cost_bits<<32 | idx) bitonic selection; exp-weighted class segment-sum;
// argmax; acc. Bandwidth-bound: 134 MB of means @ 23.3 TB/s ~ 5.8 us floor
// (means fit in 192 MB L2 -> replays run at L2 speed).
// Cross-term m.x computed with V_WMMA_F32_16X16X4_F32 (fp32-exact path).

#define C_PROTOS    262144
#define DIMS        128
#define NUM_CLASSES 1000
#define TOPL        128

#define THREADS           256
#define PROTOS_PER_BLOCK  1024
#define A_BLOCKS          (C_PROTOS / PROTOS_PER_BLOCK)   // 256
#define NKEY0             (A_BLOCKS * TOPL)               // 32768
#define B_BLOCKS          (NKEY0 / 2048)                  // 16
#define NKEY1             (B_BLOCKS * TOPL)               // 2048

typedef float v2f __attribute__((ext_vector_type(2)));
typedef float v8f __attribute__((ext_vector_type(8)));
typedef unsigned long long u64;

__device__ __forceinline__ u64 packKey(float cost, unsigned idx) {
  // cost >= 0 (clamped) -> IEEE bit pattern is order-preserving as u32.
  return ((u64)__float_as_uint(cost) << 32) | (u64)idx;
}

// Ascending bitonic sort of n (power of 2) u64 keys in LDS, whole block.
__device__ void bitonicSort(u64* keys, int n) {
  const int tid = threadIdx.x;
  for (int k = 2; k <= n; k <<= 1) {
    for (int j = k >> 1; j > 0; j >>= 1) {
      __syncthreads();
      for (int i = tid; i < n; i += THREADS) {
        int l = i ^ j;
        if (l > i) {
          u64 a = keys[i], b = keys[l];
          bool asc = ((i & k) == 0);
          if ((a > b) == asc) { keys[i] = b; keys[l] = a; }
        }
      }
    }
  }
  __syncthreads();
}

// ---------------------------------------------------------------------------
// Phase A: distances for 1024 prototypes per block (WMMA cross-term),
// in-LDS sort, emit local top-128 keys. 256 blocks keep every WGP streaming.
// ---------------------------------------------------------------------------
__global__ __launch_bounds__(THREADS)
void phaseA_cost_select(const float* __restrict__ x,
                        const float* __restrict__ means,
                        u64* __restrict__ outKeys) {
  __shared__ float xs[DIMS];
  __shared__ float dotbuf[8 * 16];             // 8 waves x 16 rows
  __shared__ u64   keys[PROTOS_PER_BLOCK];     // 8 KB

  const int tid = threadIdx.x;
  if (tid < DIMS) xs[tid] = x[tid];
  __syncthreads();

  // ||x||^2 (redundant per thread; LDS broadcast reads, negligible)
  float xnorm = 0.f;
  #pragma unroll 8
  for (int d = 0; d < DIMS; ++d) xnorm += xs[d] * xs[d];

  const int lane  = tid & 31;
  const int wave  = tid >> 5;
  const int m16   = lane & 15;      // matrix row within 16-tile
  const int khalf = lane >> 4;      // which K half-pair this lane feeds

  // Each wave owns 128 contiguous prototypes: 8 tiles of 16 rows.
  for (int t = 0; t < 8; ++t) {
    const int tileBase = wave * 128 + t * 16;                    // in-block
    const int row = blockIdx.x * PROTOS_PER_BLOCK + tileBase + m16;
    const float* rp = means + (size_t)row * DIMS;

    // Warm the next tile's lines (global_prefetch_b8); same wave reads them
    // ~one tile later, hiding L2 latency behind this tile's WMMA chain.
    if (t < 7) __builtin_prefetch(rp + 16 * DIMS, 0, 0);

    v8f acc = {0.f, 0.f, 0.f, 0.f, 0.f, 0.f, 0.f, 0.f};
    float ssq = 0.f;
    #pragma unroll 4
    for (int c = 0; c < 32; ++c) {
      const int kb = c * 4 + khalf * 2;        // A layout: 16x4 f32, 2 VGPRs
      v2f a = *(const v2f*)(rp + kb);          // global_load_b64 per lane
      v2f b; b.x = xs[kb]; b.y = xs[kb + 1];   // B[k][*] = x[k] broadcast
      ssq += a.x * a.x + a.y * a.y;            // ||m||^2 partial (free)
      // D = A(16x4) * B(4x16) + C ; every column of D = m_row . x_chunk
      acc = __builtin_amdgcn_wmma_f32_16x16x4_f32(
                false, a, false, b, (short)0, acc, false, false);
    }
    // combine the two K-halves of ||m||^2 (lanes l and l^16 share a row)
    float ssqFull = ssq + __shfl_xor(ssq, 16, 32);

    // C/D layout: VGPR r, lanes 0-15 -> M=r ; lanes 16-31 -> M=r+8.
    if (lane == 0) {
      #pragma unroll
      for (int r = 0; r < 8; ++r) dotbuf[wave * 16 + r] = acc[r];
    }
    if (lane == 16) {
      #pragma unroll
      for (int r = 0; r < 8; ++r) dotbuf[wave * 16 + 8 + r] = acc[r];
    }
    __syncthreads();
    float dot  = dotbuf[wave * 16 + m16];
    float cost = fmaxf(ssqFull - 2.f * dot + xnorm, 0.f);
    if (lane < 16) keys[tileBase + m16] = packKey(cost, (unsigned)row);
    __syncthreads();   // protect dotbuf before next tile overwrites it
  }

  bitonicSort(keys, PROTOS_PER_BLOCK);
  if (tid < TOPL) outKeys[blockIdx.x * TOPL + tid] = keys[tid];
}

// ---------------------------------------------------------------------------
// Phase B: generic 2048 -> 128 key reducer (32768 -> 2048 with 16 blocks).
// ---------------------------------------------------------------------------
__global__ __launch_bounds__(THREADS)
void phaseB_reduce(const u64* __restrict__ in, u64* __restrict__ out) {
  __shared__ u64 keys[2048];
  const int tid = threadIdx.x;
  const u64* src = in + (size_t)blockIdx.x * 2048;
  for (int i = tid; i < 2048; i += THREADS) keys[i] = src[i];
  bitonicSort(keys, 2048);
  if (tid < TOPL) out[blockIdx.x * TOPL + tid] = keys[tid];
}

// ---------------------------------------------------------------------------
// Phase C: 2048 -> top-128, then exp weights, class segment-sum, argmax, acc.
// ---------------------------------------------------------------------------
__global__ __launch_bounds__(THREADS)
void phaseC_final(const u64* __restrict__ in,
                  const int* __restrict__ clusterClasses,
                  const int* __restrict__ target,
                  const float* __restrict__ variance,
                  float* __restrict__ out) {
  __shared__ u64   keys[NKEY1];
  __shared__ float sProb[NUM_CLASSES];
  __shared__ float sDen;
  __shared__ float bp[THREADS];
  __shared__ int   bc[THREADS];

  const int tid = threadIdx.x;
  for (int i = tid; i < NKEY1; i += THREADS) keys[i] = in[i];
  bitonicSort(keys, NKEY1);

  for (int i = tid; i < NUM_CLASSES; i += THREADS) sProb[i] = 0.f;
  if (tid == 0) sDen = 0.f;
  __syncthreads();

  const float v  = variance[0];
  const float vn = -1.f / (2.f * v * v);
  if (tid < TOPL) {
    u64 k = keys[tid];
    float    cost = __uint_as_float((unsigned)(k >> 32));
    unsigned idx  = (unsigned)(k & 0xffffffffu);
    int cls = clusterClasses[idx];
    float w = expf(vn * cost);
    atomicAdd(&sProb[cls], w);   // ds_add_f32
    atomicAdd(&sDen, w);
  }
  __syncthreads();

  const float den = sDen + 1e-8f;
  float bestP = -1.f; int bestC = 0;
  for (int c = tid; c < NUM_CLASSES; c += THREADS) {
    float p = sProb[c] / den;
    if (p > bestP) { bestP = p; bestC = c; }   // strided -> keeps lowest c
  }
  bp[tid] = bestP; bc[tid] = bestC;
  __syncthreads();
  if (tid == 0) {
    float P = bp[0]; int Cc = bc[0];
    for (int t2 = 1; t2 < THREADS; ++t2) {
      if (bp[t2] > P || (bp[t2] == P && bc[t2] < Cc)) { P = bp[t2]; Cc = bc[t2]; }
    }
    out[0] = 0.f;
    out[1] = 0.f;
    out[2] = (float)Cc;                          // pred
    out[3] = (Cc == target[0]) ? 1.f : 0.f;      // acc
  }
}

// ---------------------------------------------------------------------------
extern "C" void kernel_launch(void* const* d_in, const int* in_sizes, int n_in,
                              void* d_out, int out_size, void* d_ws, size_t ws_size,
                              hipStream_t stream) {
  const float* x        = (const float*)d_in[0];
  const int*   target   = (const int*)  d_in[1];
  const float* means    = (const float*)d_in[2];
  const int*   classes  = (const int*)  d_in[3];
  const float* variance = (const float*)d_in[4];
  float*       out      = (float*)d_out;

  u64* ws0 = (u64*)d_ws;          // 32768 keys (256 KB)
  u64* ws1 = ws0 + NKEY0;         // 2048 keys (16 KB)

  phaseA_cost_select<<<A_BLOCKS, THREADS, 0, stream>>>(x, means, ws0);
  phaseB_reduce<<<B_BLOCKS, THREADS, 0, stream>>>(ws0, ws1);
  phaseC_final<<<1, THREADS, 0, stream>>>(ws1, classes, target, variance, out);
}